// HeteroVGAEEncoder_51917564674735
// MI455X (gfx1250) — compile-verified
//
#include <hip/hip_runtime.h>

typedef __attribute__((ext_vector_type(2))) float v2f;
typedef __attribute__((ext_vector_type(8))) float v8f;

// ---------------------------------------------------------------------------
// Utility fill kernels
// ---------------------------------------------------------------------------
__global__ void fill_zero_f32(float* __restrict__ p, int n) {
  int i = blockIdx.x * blockDim.x + threadIdx.x;
  if (i < n) p[i] = 0.0f;
}

// out_d[n] = b2_d ; out_g[n] = b2_g  (layer-2 bias seed; isolated nodes keep it)
__global__ void init_out_kernel(float* __restrict__ out_d, int nd,
                                float* __restrict__ out_g, int ng,
                                const float* __restrict__ b2d,
                                const float* __restrict__ b2g) {
  int i = blockIdx.x * blockDim.x + threadIdx.x;
  if (i < nd) out_d[i] = b2d[0];
  if (i < ng) out_g[i] = b2g[0];
}

// ---------------------------------------------------------------------------
// Degree count (in-degree over dst) accumulated as f32 (atomics hit L2-resident
// 400KB array -> cheap on 192MB L2)
// ---------------------------------------------------------------------------
__global__ void degree_kernel(const long long* __restrict__ ei, // [2,E]
                              float* __restrict__ deg, int E) {
  int e = blockIdx.x * blockDim.x + threadIdx.x;
  if (e < E) {
    int d = (int)ei[(long long)E + e];
    atomicAdd(&deg[d], 1.0f);
  }
}

// deg -> dinv = deg>0 ? rsqrt(deg) : 0   (in place)
__global__ void dinv_kernel(float* __restrict__ deg, int n) {
  int i = blockIdx.x * blockDim.x + threadIdx.x;
  if (i < n) {
    float d = deg[i];
    deg[i] = (d > 0.0f) ? rsqrtf(d) : 0.0f;
  }
}

// ---------------------------------------------------------------------------
// Layer-1 scalar aggregation: agg[dst] += dinv[src]*dinv[dst]*x[src]
// (1 channel for disease, 2 channels for gene)
// ---------------------------------------------------------------------------
__global__ void scatter1_c1_kernel(const long long* __restrict__ ei,
                                   const float* __restrict__ dinv,
                                   const float* __restrict__ x,   // [N,1]
                                   float* __restrict__ agg,       // [N]
                                   int E) {
  int e = blockIdx.x * blockDim.x + threadIdx.x;
  if (e < E) {
    int s = (int)ei[e];
    int d = (int)ei[(long long)E + e];
    float nr = dinv[s] * dinv[d];
    atomicAdd(&agg[d], nr * x[s]);
  }
}

__global__ void scatter1_c2_kernel(const long long* __restrict__ ei,
                                   const float* __restrict__ dinv,
                                   const float* __restrict__ x,   // [N,2]
                                   float* __restrict__ agg,       // [N,2]
                                   int E) {
  int e = blockIdx.x * blockDim.x + threadIdx.x;
  if (e < E) {
    int s = (int)ei[e];
    int d = (int)ei[(long long)E + e];
    float nr = dinv[s] * dinv[d];
    atomicAdd(&agg[2 * d + 0], nr * x[2 * s + 0]);
    atomicAdd(&agg[2 * d + 1], nr * x[2 * s + 1]);
  }
}

// ---------------------------------------------------------------------------
// Per-node MLP via WMMA f32 16x16x4 (exact f32 math):
//   s[n] = sum_k relu(agg[n,:]·W1[:,k] + b1[k]) * W2[k]
// One wave handles a tile of 16 nodes. H-tile (16x32) times W2 (32x1) is done
// as 8 chained V_WMMA_F32_16X16X4_F32 ops with W2 broadcast across all 16
// columns of B, so every column of D carries s; lanes 0 and 16 extract rows
// 0-7 and 8-15 from the 8 accumulator VGPRs.
// A 16x4 f32 layout (ISA 7.12.2): lanes 0-15 -> K=0(V0),K=1(V1);
//                                  lanes 16-31 -> K=2(V0),K=3(V1).
// EXEC kept all-ones: tail tiles use clamped loads, masking only at store.
// ---------------------------------------------------------------------------
__global__ void mlp_wmma_kernel(const float* __restrict__ agg,  // [N,in_dim]
                                const float* __restrict__ W1,   // [in_dim,32]
                                const float* __restrict__ b1,   // [32]
                                const float* __restrict__ W2,   // [32]
                                float* __restrict__ s,          // [N]
                                int N, int in_dim) {
  int lane = threadIdx.x & 31;
  int wave = (blockIdx.x * blockDim.x + threadIdx.x) >> 5;
  int tile = wave;                       // 16 nodes per wave
  int m = lane & 15;
  int node = tile * 16 + m;
  int nodeC = node < N ? node : (N - 1); // clamp: keep EXEC full for WMMA

  float a0 = agg[nodeC * in_dim + 0];
  float a1 = (in_dim == 2) ? agg[nodeC * in_dim + 1] : 0.0f;

  int khalf = (lane >> 4) << 1;          // 0 for lanes 0-15, 2 for lanes 16-31

  v8f c = {0.f, 0.f, 0.f, 0.f, 0.f, 0.f, 0.f, 0.f};
#pragma unroll
  for (int kk = 0; kk < 8; ++kk) {
    int k0 = kk * 4 + khalf;
    int k1 = k0 + 1;
    // H(m, k) = relu(agg·W1[:,k] + b1[k])
    float h0 = a0 * W1[k0] + b1[k0];
    float h1 = a0 * W1[k1] + b1[k1];
    if (in_dim == 2) {
      h0 += a1 * W1[32 + k0];
      h1 += a1 * W1[32 + k1];
    }
    h0 = h0 > 0.0f ? h0 : 0.0f;
    h1 = h1 > 0.0f ? h1 : 0.0f;
    v2f A; A[0] = h0; A[1] = h1;
    // B 4x16: broadcast W2 chunk across all columns (value depends only on K)
    v2f B; B[0] = W2[k0]; B[1] = W2[k1];
    c = __builtin_amdgcn_wmma_f32_16x16x4_f32(
        /*neg_a=*/false, A, /*neg_b=*/false, B,
        /*c_mod=*/(short)0, c, /*reuse_a=*/false, /*reuse_b=*/false);
  }

  // D layout: VGPR j, lanes 0-15: M=j; lanes 16-31: M=8+j (all columns equal)
  if (lane == 0 || lane == 16) {
    int base = tile * 16 + ((lane >> 4) << 3);
#pragma unroll
    for (int j = 0; j < 8; ++j) {
      int n = base + j;
      if (n < N) s[n] = c[j];
    }
  }
}

// ---------------------------------------------------------------------------
// Layer-2 scalar aggregation into output (pre-seeded with b2):
//   out[dst] += dinv[src]*dinv[dst]*s[src]
// ---------------------------------------------------------------------------
__global__ void scatter2_kernel(const long long* __restrict__ ei,
                                const float* __restrict__ dinv,
                                const float* __restrict__ s,
                                float* __restrict__ out, int E) {
  int e = blockIdx.x * blockDim.x + threadIdx.x;
  if (e < E) {
    int sN = (int)ei[e];
    int d = (int)ei[(long long)E + e];
    atomicAdd(&out[d], dinv[sN] * dinv[d] * s[sN]);
  }
}

// ---------------------------------------------------------------------------
// Launcher
// ---------------------------------------------------------------------------
extern "C" void kernel_launch(void* const* d_in, const int* in_sizes, int n_in,
                              void* d_out, int out_size, void* d_ws, size_t ws_size,
                              hipStream_t stream) {
  const float*     x_d  = (const float*)d_in[0];       // [N_D,1]
  const float*     x_g  = (const float*)d_in[1];       // [N_G,2]
  const long long* ei_d = (const long long*)d_in[2];   // [2,E]
  const long long* ei_g = (const long long*)d_in[3];   // [2,E]
  const float*     W1d  = (const float*)d_in[4];       // [1,32]
  const float*     b1d  = (const float*)d_in[5];       // [32]
  const float*     W1g  = (const float*)d_in[6];       // [2,32]
  const float*     b1g  = (const float*)d_in[7];       // [32]
  const float*     W2d  = (const float*)d_in[8];       // [32,1]
  const float*     b2d  = (const float*)d_in[9];       // [1]
  const float*     W2g  = (const float*)d_in[10];      // [32,1]
  const float*     b2g  = (const float*)d_in[11];      // [1]

  const int N_D = in_sizes[0];
  const int N_G = in_sizes[1] / 2;
  const int E_D = in_sizes[2] / 2;
  const int E_G = in_sizes[3] / 2;

  float* out_d = (float*)d_out;          // [N_D]
  float* out_g = (float*)d_out + N_D;    // [N_G]

  // Workspace layout (f32): dinv_d | dinv_g | agg_d | agg_g(2ch) | s_d | s_g
  float* dinv_d = (float*)d_ws;
  float* dinv_g = dinv_d + N_D;
  float* agg_d  = dinv_g + N_G;
  float* agg_g  = agg_d + N_D;           // [N_G,2]
  float* s_d    = agg_g + 2 * N_G;
  float* s_g    = s_d + N_D;
  const int ws_f32 = N_D + N_G + N_D + 2 * N_G + N_D + N_G;

  const int T = 256;
  auto blocks = [](int n, int t) { return (n + t - 1) / t; };

  // 0) zero scratch, seed output with layer-2 bias
  fill_zero_f32<<<blocks(ws_f32, T), T, 0, stream>>>((float*)d_ws, ws_f32);
  {
    int n = N_D > N_G ? N_D : N_G;
    init_out_kernel<<<blocks(n, T), T, 0, stream>>>(out_d, N_D, out_g, N_G, b2d, b2g);
  }

  // 1) degrees -> dinv
  degree_kernel<<<blocks(E_D, T), T, 0, stream>>>(ei_d, dinv_d, E_D);
  degree_kernel<<<blocks(E_G, T), T, 0, stream>>>(ei_g, dinv_g, E_G);
  dinv_kernel<<<blocks(N_D, T), T, 0, stream>>>(dinv_d, N_D);
  dinv_kernel<<<blocks(N_G, T), T, 0, stream>>>(dinv_g, N_G);

  // 2) layer-1 scalar aggregation (1 / 2 channels per edge)
  scatter1_c1_kernel<<<blocks(E_D, T), T, 0, stream>>>(ei_d, dinv_d, x_d, agg_d, E_D);
  scatter1_c2_kernel<<<blocks(E_G, T), T, 0, stream>>>(ei_g, dinv_g, x_g, agg_g, E_G);

  // 3) per-node MLP via WMMA f32 16x16x4: s[n] = relu(agg·W1 + b1) · W2
  {
    int tiles_d = (N_D + 15) / 16;                 // one wave per tile
    int tiles_g = (N_G + 15) / 16;
    int wavesPerBlock = T / 32;
    mlp_wmma_kernel<<<(tiles_d + wavesPerBlock - 1) / wavesPerBlock, T, 0, stream>>>(
        agg_d, W1d, b1d, W2d, s_d, N_D, 1);
    mlp_wmma_kernel<<<(tiles_g + wavesPerBlock - 1) / wavesPerBlock, T, 0, stream>>>(
        agg_g, W1g, b1g, W2g, s_g, N_G, 2);
  }

  // 4) layer-2 scalar aggregation into the bias-seeded outputs
  scatter2_kernel<<<blocks(E_D, T), T, 0, stream>>>(ei_d, dinv_d, s_d, out_d, E_D);
  scatter2_kernel<<<blocks(E_G, T), T, 0, stream>>>(ei_g, dinv_g, s_g, out_g, E_G);
}